// FactorizedConvolution_16707422781943
// MI455X (gfx1250) — compile-verified
//
#include <hip/hip_runtime.h>
#include <hip/hip_bf16.h>
#include <math.h>

// N=50000, E=800000, U=64, R=8, A=4, NPATH=5
// out layout (N,256): [0:64] = scalar channel u ; [64 + u*3 + i] = vector channel (u,i)

typedef float v2f __attribute__((ext_vector_type(2)));
typedef float v8f __attribute__((ext_vector_type(8)));

__device__ __forceinline__ v8f wmma_f32(v2f a, v2f b, v8f c) {
  // V_WMMA_F32_16X16X4_F32 : C(16x16,f32) += A(16x4,f32) * B(4x16,f32)
  return __builtin_amdgcn_wmma_f32_16x16x4_f32(
      /*neg_a=*/false, a, /*neg_b=*/false, b,
      /*c_mod=*/(short)0, c, /*reuse_a=*/false, /*reuse_b=*/false);
}

// ---------------------------------------------------------------------------
// Node kernel: per 16-node tile compute
//   y0 = x0 @ Wl0 * 1/sqrt(U)                 -> ws
//   y1_i = x1_i @ Wl1 * 1/sqrt(U)             -> ws
//   sc0 = (x0 (x) na) @ sc_w0 * 1/sqrt(U*A)   -> out[:, 0:64]
//   sc1_i = (x1_i (x) na) @ sc_w1 * 1/sqrt(U*A) -> out[:, 64+u*3+i]
// 4 waves/block, wave w owns output columns [16w, 16w+16).
// A layout (16x4 f32): lane<16: row=lane, K={0,1}; lane>=16: row=lane-16, K={2,3}
// B layout (4x16 f32): lane<16: col=lane, K={0,1}; lane>=16: col=lane-16, K={2,3}
// C layout: vgpr r -> row r (lanes 0-15) / row r+8 (lanes 16-31), col = lane&15
// ---------------------------------------------------------------------------
__global__ __launch_bounds__(128) void node_kernel(
    const float* __restrict__ x, const float* __restrict__ na,
    const float* __restrict__ Wl0, const float* __restrict__ Wl1,
    const float* __restrict__ sw0, const float* __restrict__ sw1,
    float* __restrict__ y0ws, float* __restrict__ y1ws,
    float* __restrict__ out, int N)
{
  const int lane  = threadIdx.x & 31;
  const int wv    = threadIdx.x >> 5;
  const int node0 = blockIdx.x * 16;
  const int nb    = wv * 16;          // column tile base
  const int m     = lane & 15;        // A-row / C-col within tile
  const int hi    = lane >> 4;        // K-half selector
  int rowi = node0 + m; if (rowi > N - 1) rowi = N - 1;
  const float* xr  = x  + (size_t)rowi * 256;   // x0 = xr[0:64], x1[u][i] = xr[64+u*3+i]
  const float* nar = na + (size_t)rowi * 4;

  v8f y0a = {}, y1a0 = {}, y1a1 = {}, y1a2 = {};
  v8f s0a = {}, s1a0 = {}, s1a1 = {}, s1a2 = {};

  // ---- linear GEMMs: K = 64 ----
  #pragma unroll 4
  for (int k0 = 0; k0 < 64; k0 += 4) {
    const int ka = k0 + hi * 2;
    v2f a, b;
    a[0] = xr[ka]; a[1] = xr[ka + 1];
    b[0] = Wl0[ka * 64 + nb + m]; b[1] = Wl0[(ka + 1) * 64 + nb + m];
    y0a = wmma_f32(a, b, y0a);
    b[0] = Wl1[ka * 64 + nb + m]; b[1] = Wl1[(ka + 1) * 64 + nb + m];
    a[0] = xr[64 + ka * 3 + 0]; a[1] = xr[64 + (ka + 1) * 3 + 0];
    y1a0 = wmma_f32(a, b, y1a0);
    a[0] = xr[64 + ka * 3 + 1]; a[1] = xr[64 + (ka + 1) * 3 + 1];
    y1a1 = wmma_f32(a, b, y1a1);
    a[0] = xr[64 + ka * 3 + 2]; a[1] = xr[64 + (ka + 1) * 3 + 2];
    y1a2 = wmma_f32(a, b, y1a2);
  }

  // ---- self-connection GEMMs: K = U*A = 256, A built on the fly ----
  const float na0 = nar[hi * 2 + 0];
  const float na1 = nar[hi * 2 + 1];
  #pragma unroll 2
  for (int k0 = 0; k0 < 256; k0 += 4) {
    const int ka = k0 + hi * 2;       // this lane's first K; u is constant in the step
    const int u  = k0 >> 2;
    v2f a, b;
    const float x0v = xr[u];
    b[0] = sw0[ka * 64 + nb + m]; b[1] = sw0[(ka + 1) * 64 + nb + m];
    a[0] = x0v * na0; a[1] = x0v * na1;
    s0a = wmma_f32(a, b, s0a);
    b[0] = sw1[ka * 64 + nb + m]; b[1] = sw1[(ka + 1) * 64 + nb + m];
    const float x1x = xr[64 + u * 3 + 0];
    a[0] = x1x * na0; a[1] = x1x * na1;
    s1a0 = wmma_f32(a, b, s1a0);
    const float x1y = xr[64 + u * 3 + 1];
    a[0] = x1y * na0; a[1] = x1y * na1;
    s1a1 = wmma_f32(a, b, s1a1);
    const float x1z = xr[64 + u * 3 + 2];
    a[0] = x1z * na0; a[1] = x1z * na1;
    s1a2 = wmma_f32(a, b, s1a2);
  }

  const float inv_sqrtU = 0.125f;     // 1/sqrt(64)
  const float sc_norm   = 0.0625f;    // 1/sqrt(64*4)
  #pragma unroll
  for (int r = 0; r < 8; ++r) {
    const int mr = node0 + r + hi * 8;
    if (mr < N) {
      const int col = nb + m;
      y0ws[(size_t)mr * 64 + col] = y0a[r] * inv_sqrtU;
      float* y1p = y1ws + (size_t)mr * 192 + col * 3;
      y1p[0] = y1a0[r] * inv_sqrtU;
      y1p[1] = y1a1[r] * inv_sqrtU;
      y1p[2] = y1a2[r] * inv_sqrtU;
      float* op = out + (size_t)mr * 256;
      op[col]               = s0a[r] * sc_norm;
      op[64 + col * 3 + 0]  = s1a0[r] * sc_norm;
      op[64 + col * 3 + 1]  = s1a1[r] * sc_norm;
      op[64 + col * 3 + 2]  = s1a2[r] * sc_norm;
    }
  }
}

// ---------------------------------------------------------------------------
// Edge kernel: one wave32 per edge, 2 channels per lane.
// Radial MLP in registers, gather y[src] (L2-resident), tensor-product math,
// f32 atomic scatter into out[dst] (L2-resident).
// ---------------------------------------------------------------------------
__global__ __launch_bounds__(256) void edge_kernel(
    const float* __restrict__ er, const float* __restrict__ es,
    const int* __restrict__ eidx,
    const float* __restrict__ fw1, const float* __restrict__ fw2,
    const float* __restrict__ y0ws, const float* __restrict__ y1ws,
    float* __restrict__ out, int E)
{
  const int e = blockIdx.x * 8 + (threadIdx.x >> 5);
  if (e >= E) return;
  const int lane = threadIdx.x & 31;

  float r[8];
  #pragma unroll
  for (int k = 0; k < 8; ++k) r[k] = er[(size_t)e * 8 + k];

  const float is8 = 0.35355339059327373f;   // 1/sqrt(8)
  float h[8];
  #pragma unroll
  for (int j = 0; j < 8; ++j) {
    float acc = 0.f;
    #pragma unroll
    for (int k = 0; k < 8; ++k) acc = fmaf(r[k], fw1[k * 8 + j], acc);
    acc *= is8;
    // stable softplus(acc) - log(2)
    float sp = fmaxf(acc, 0.f) + log1pf(__expf(-fabsf(acc)));
    h[j] = sp - 0.6931471805599453f;
  }

  const int src = eidx[e];
  const int dst = eidx[E + e];
  const float e0  = es[(size_t)e * 4 + 0];
  const float e1x = es[(size_t)e * 4 + 1];
  const float e1y = es[(size_t)e * 4 + 2];
  const float e1z = es[(size_t)e * 4 + 3];
  const float* s0b = y0ws + (size_t)src * 64;
  const float* s1b = y1ws + (size_t)src * 192;
  float* ob = out + (size_t)dst * 256;

  const float is3 = 0.5773502691896258f;    // 1/sqrt(3)
  const float is2 = 0.7071067811865476f;    // 1/sqrt(2)
  const float agg = 0.25f;                  // 1/sqrt(AVG_NEIGH)

  #pragma unroll
  for (int cc = 0; cc < 2; ++cc) {
    const int c = lane + cc * 32;
    float w0 = 0.f, w1 = 0.f, w2 = 0.f, w3 = 0.f, w4 = 0.f;
    #pragma unroll
    for (int j = 0; j < 8; ++j) {
      const float hh = h[j];
      const float* f2 = fw2 + j * 320 + c;
      w0 = fmaf(hh, f2[0],   w0);
      w1 = fmaf(hh, f2[64],  w1);
      w2 = fmaf(hh, f2[128], w2);
      w3 = fmaf(hh, f2[192], w3);
      w4 = fmaf(hh, f2[256], w4);
    }
    w0 *= is8; w1 *= is8; w2 *= is8; w3 *= is8; w4 *= is8;

    const float s0  = s0b[c];
    const float s1x = s1b[c * 3 + 0];
    const float s1y = s1b[c * 3 + 1];
    const float s1z = s1b[c * 3 + 2];

    const float dotv = s1x * e1x + s1y * e1y + s1z * e1z;
    const float cx = s1y * e1z - s1z * e1y;
    const float cy = s1z * e1x - s1x * e1z;
    const float cz = s1x * e1y - s1y * e1x;

    const float o0 = (w0 * s0 * e0 + w1 * dotv * is3) * is2 * agg;
    const float k4 = w4 * is2;
    const float k2 = w2 * s0;
    const float k3 = w3 * e0;
    const float o1x = (k2 * e1x + k3 * s1x + k4 * cx) * is3 * agg;
    const float o1y = (k2 * e1y + k3 * s1y + k4 * cy) * is3 * agg;
    const float o1z = (k2 * e1z + k3 * s1z + k4 * cz) * is3 * agg;

    atomicAdd(ob + c, o0);
    atomicAdd(ob + 64 + c * 3 + 0, o1x);
    atomicAdd(ob + 64 + c * 3 + 1, o1y);
    atomicAdd(ob + 64 + c * 3 + 2, o1z);
  }
}

extern "C" void kernel_launch(void* const* d_in, const int* in_sizes, int n_in,
                              void* d_out, int out_size, void* d_ws, size_t ws_size,
                              hipStream_t stream) {
  const float* x   = (const float*)d_in[0];
  const float* na  = (const float*)d_in[1];
  const float* er  = (const float*)d_in[2];
  const float* es  = (const float*)d_in[3];
  const int*   ei  = (const int*)d_in[4];
  const float* Wl0 = (const float*)d_in[5];
  const float* Wl1 = (const float*)d_in[6];
  const float* fw1 = (const float*)d_in[7];
  const float* fw2 = (const float*)d_in[8];
  const float* sw0 = (const float*)d_in[9];
  const float* sw1 = (const float*)d_in[10];

  const int N = in_sizes[0] / 256;
  const int E = in_sizes[2] / 8;

  float* out  = (float*)d_out;
  float* y0ws = (float*)d_ws;                 // N*64 floats
  float* y1ws = y0ws + (size_t)N * 64;        // N*192 floats  (total 51.2 MB)

  const int ntiles = (N + 15) / 16;
  node_kernel<<<ntiles, 128, 0, stream>>>(x, na, Wl0, Wl1, sw0, sw1,
                                          y0ws, y1ws, out, N);

  const int eblocks = (E + 7) / 8;
  edge_kernel<<<eblocks, 256, 0, stream>>>(er, es, ei, fw1, fw2,
                                           y0ws, y1ws, out, E);
}